// LSTMTranslationModel_22179211116605
// MI455X (gfx1250) — compile-verified
//
#include <hip/hip_runtime.h>
#include <hip/hip_bf16.h>
#include <math.h>
#include <stdint.h>

// ---------------------------------------------------------------------------
// LSTM translation model, CDNA5 (gfx1250)
//  - f32 WMMA (V_WMMA_F32_16X16X4_F32) for all GEMMs (reference is f32; the
//    kernel is HBM-bound on the 262MB output, so f32 matrix path costs nothing)
//  - Tensor Data Mover (tensor_load_to_lds) stages W_out tiles into LDS with
//    hardware padding for bank-conflict-free reads; double-buffered on TENSORcnt
// B=4 S=128 -> N=512, D=512, E=256, H=512 (4H=2048), V=32000, K=4 steps
// ---------------------------------------------------------------------------

typedef float v2f __attribute__((ext_vector_type(2)));
typedef float v8f __attribute__((ext_vector_type(8)));
typedef unsigned int u32x4 __attribute__((ext_vector_type(4)));
typedef int i32x4 __attribute__((ext_vector_type(4)));
typedef int i32x8 __attribute__((ext_vector_type(8)));

#define N_ROWS 512
#define DDIM   512
#define EDIM   256
#define HDIM   512
#define G4     2048
#define VOCAB  32000
#define KSTEPS 4
#define OOV_ID 1

#if __has_builtin(__builtin_amdgcn_tensor_load_to_lds) && __has_builtin(__builtin_amdgcn_s_wait_tensorcnt)
#define HAVE_TDM 1
#else
#define HAVE_TDM 0
#endif

__device__ __forceinline__ v8f wmma4(v2f a, v2f b, v8f c) {
  // D = A(16x4,f32) * B(4x16,f32) + C(16x16,f32)
  return __builtin_amdgcn_wmma_f32_16x16x4_f32(
      /*neg_a=*/false, a, /*neg_b=*/false, b,
      /*c_mod=*/(short)0, c, /*reuse_a=*/false, /*reuse_b=*/false);
}

__device__ __forceinline__ float sigmoidf_(float x) {
  return 1.0f / (1.0f + __expf(-x));
}

// ------------------------------------------------------------------ utilities
__global__ void zero_kernel(float* __restrict__ p, int n) {
  int t = blockIdx.x * blockDim.x + threadIdx.x;
  if (t < n) p[t] = 0.0f;
}

// step 0: last = one-hot(OOV_ID) -> emb_last row = emb_table[OOV_ID]
__global__ void emb_init_kernel(const float* __restrict__ emb_table,
                                float* __restrict__ embl) {
  int t = blockIdx.x * 256 + threadIdx.x;       // N*E = 131072
  embl[t] = emb_table[OOV_ID * EDIM + (t & (EDIM - 1))];
}

// ---------------------------------------------------------------- emb bag
// emb_last(N,E) = exp(logp_prev)(N,V) @ emb_table(V,E)
// block = 256 thr (8 waves). grid = (N/16, E/128). A-chunk (16xV-slice) in LDS.
__global__ __launch_bounds__(256)
void emb_bag_kernel(const float* __restrict__ out_prev,   // full d_out
                    const float* __restrict__ emb_table,
                    float* __restrict__ embl, int step_prev) {
  __shared__ float As[16 * 132];                 // pad 128->132: conflict-free
  const int lane = threadIdx.x & 31;
  const int wave = threadIdx.x >> 5;
  const int m0   = blockIdx.x * 16;
  const int e    = blockIdx.y * 128 + wave * 16 + (lane & 15);
  const int kh   = (lane >> 4) * 2;              // K-half: 0 or 2
  const int ml   = lane & 15;

  v8f acc = {};
  for (int v0 = 0; v0 < VOCAB; v0 += 128) {
    __syncthreads();
    for (int t = threadIdx.x; t < 16 * 128; t += 256) {
      int rm = t >> 7, cc = t & 127;
      size_t idx = ((size_t)((m0 + rm) * KSTEPS + step_prev)) * VOCAB + v0 + cc;
      As[rm * 132 + cc] = __expf(out_prev[idx]);
    }
    __syncthreads();
    for (int kk = 0; kk < 128; kk += 4) {
      v2f a, b;
      a.x = As[ml * 132 + kk + kh];
      a.y = As[ml * 132 + kk + kh + 1];
      const float* bp = emb_table + (size_t)(v0 + kk + kh) * EDIM + e;
      b.x = bp[0];
      b.y = bp[EDIM];
      acc = wmma4(a, b, acc);
    }
  }
#pragma unroll
  for (int r = 0; r < 8; ++r) {
    int row = m0 + r + ((lane >> 4) << 3);
    embl[row * EDIM + e] = acc[r];
  }
}

// ---------------------------------------------------------------- gates
// gates(N,4H) = [x|embl](N,768) @ W_ih^T + h(N,512) @ W_hh^T + b_ih + b_hh
// block = 256 thr (8 waves, one 16x16 tile each). grid = (N/16, 2048/128)
__global__ __launch_bounds__(256)
void gates_kernel(const float* __restrict__ x,      // N x D
                  const float* __restrict__ embl,   // N x E
                  const float* __restrict__ h,      // N x H
                  const float* __restrict__ W_ih,   // 4H x (D+E)
                  const float* __restrict__ b_ih,
                  const float* __restrict__ W_hh,   // 4H x H
                  const float* __restrict__ b_hh,
                  float* __restrict__ gates) {      // N x 4H
  const int lane = threadIdx.x & 31;
  const int wave = threadIdx.x >> 5;
  const int m0   = blockIdx.x * 16;
  const int j    = (blockIdx.y * 8 + wave) * 16 + (lane & 15);
  const int kh   = (lane >> 4) * 2;
  const int ml   = lane & 15;
  const int m    = m0 + ml;

  const float bias = b_ih[j] + b_hh[j];
  v8f acc;
#pragma unroll
  for (int r = 0; r < 8; ++r) acc[r] = bias;

  // K segment 1: x (p = 0..511 of D+E)
  for (int p = 0; p < DDIM; p += 4) {
    v2f a, b;
    a.x = x[m * DDIM + p + kh];
    a.y = x[m * DDIM + p + kh + 1];
    b.x = W_ih[j * (DDIM + EDIM) + p + kh];
    b.y = W_ih[j * (DDIM + EDIM) + p + kh + 1];
    acc = wmma4(a, b, acc);
  }
  // K segment 2: emb_last (p = 512..767)
  for (int p = 0; p < EDIM; p += 4) {
    v2f a, b;
    a.x = embl[m * EDIM + p + kh];
    a.y = embl[m * EDIM + p + kh + 1];
    b.x = W_ih[j * (DDIM + EDIM) + DDIM + p + kh];
    b.y = W_ih[j * (DDIM + EDIM) + DDIM + p + kh + 1];
    acc = wmma4(a, b, acc);
  }
  // K segment 3: h @ W_hh^T
  for (int p = 0; p < HDIM; p += 4) {
    v2f a, b;
    a.x = h[m * HDIM + p + kh];
    a.y = h[m * HDIM + p + kh + 1];
    b.x = W_hh[j * HDIM + p + kh];
    b.y = W_hh[j * HDIM + p + kh + 1];
    acc = wmma4(a, b, acc);
  }
#pragma unroll
  for (int r = 0; r < 8; ++r) {
    int row = m0 + r + ((lane >> 4) << 3);
    gates[row * G4 + j] = acc[r];
  }
}

// ---------------------------------------------------------------- LSTM cell
__global__ void cell_kernel(const float* __restrict__ gates,
                            float* __restrict__ h, float* __restrict__ c) {
  int t = blockIdx.x * blockDim.x + threadIdx.x;   // N*H = 262144
  int n = t >> 9, j = t & (HDIM - 1);
  const float* g = gates + n * G4;
  float i  = sigmoidf_(g[j]);
  float f  = sigmoidf_(g[HDIM + j]);
  float gg = tanhf(g[2 * HDIM + j]);
  float o  = sigmoidf_(g[3 * HDIM + j]);
  float cn = f * c[t] + i * gg;
  float hn = o * tanhf(cn);
  c[t] = cn;
  h[t] = hn;
}

// ---------------------------------------------------------------- logits
// logits(N,V) = h(N,512) @ W_out^T + b_out, written into d_out slice `step`.
// block = 256 thr (8 waves). grid = (N/16, V/128).
//  A: h row-block staged in LDS (16x516, conflict-free).
//  B: W_out tiles DMA'd by the Tensor Data Mover into LDS, 128 rows x 16 K
//     per chunk, HW-padded +4 DWORDs per 16 (row stride 20 -> conflict-free),
//     double-buffered on TENSORcnt.
#define BCHUNK 16
#define BPAD   20   // 16 + 4 pad DWORDs inserted by TDM

__global__ __launch_bounds__(256)
void logits_kernel(const float* __restrict__ h,
                   const float* __restrict__ W_out,   // V x H
                   const float* __restrict__ b_out,
                   float* __restrict__ out, int step) {
  __shared__ float As[16 * 516];                 // 33.0 KB
#if HAVE_TDM
  __shared__ float Bs[2][128 * BPAD];            // 20.0 KB
#endif
  const int m0 = blockIdx.x * 16;
  const int jb = blockIdx.y * 128;

  for (int t = threadIdx.x; t < 16 * 512; t += 256) {
    int rm = t >> 9, cc = t & 511;
    As[rm * 516 + cc] = h[(m0 + rm) * HDIM + cc];
  }

  const int lane = threadIdx.x & 31;
  const int wave = threadIdx.x >> 5;
  const int jl   = wave * 16 + (lane & 15);      // block-local column 0..127
  const int j    = jb + jl;
  const int kh   = (lane >> 4) * 2;
  const int ml   = lane & 15;

  const float bias = b_out[j];
  v8f acc;
#pragma unroll
  for (int r = 0; r < 8; ++r) acc[r] = bias;

#if HAVE_TDM
  const bool issuer = (wave == 0);               // TDM ignores EXEC: issue from
                                                 // exactly one wave (uniform branch)
  const unsigned lds_base = (unsigned)(uintptr_t)(&Bs[0][0]);
  const unsigned long long ga_base =
      (unsigned long long)(uintptr_t)W_out + (unsigned long long)jb * HDIM * 4ull;

  auto tdm_issue = [&](int ch, int buf) {
    const unsigned long long ga = ga_base + (unsigned long long)(ch * BCHUNK) * 4ull;
    u32x4 g0;
    g0.x = 1u;                                   // count=1, user-mode, no gather
    g0.y = lds_base + (unsigned)buf * (128u * BPAD * 4u);  // lds_addr (bytes)
    g0.z = (unsigned)(ga & 0xffffffffull);       // global_addr[31:0]
    g0.w = (unsigned)(ga >> 32) | (2u << 30);    // global_addr[56:32] | type=2
    i32x8 g1;
    g1[0] = (2 << 16)        // data_size = 4B
          | (1 << 20)        // pad_enable
          | (3 << 22)        // pad_interval: every 16 DWORDs
          | (3 << 25);       // pad_amount:  4 DWORDs  -> LDS row stride 20
    g1[1] = (int)(512u << 16);     // [15:0] atomic_barrier=0, [31:16] tensor_dim0 lo (512)
    g1[2] = (int)(32000u << 16);   // [15:0] tensor_dim0 hi=0, [31:16] tensor_dim1 lo (32000)
    g1[3] = (int)((unsigned)BCHUNK << 16); // [15:0] tensor_dim1 hi=0, [31:16] tile_dim0
    g1[4] = 128;                   // [15:0] tile_dim1=128 rows, [31:16] tile_dim2=0
    g1[5] = 512;                   // tensor_dim0_stride lo (H)
    g1[6] = 0;                     // stride hi / dim1_stride (unused, 2D tile)
    g1[7] = 0;
    const i32x4 gz4 = {0, 0, 0, 0};
    const i32x8 gz8 = {0, 0, 0, 0, 0, 0, 0, 0};
    // 6-arg form (clang-23 / therock-10.0): (g0, g1, g2, g3, g4, cpol)
    __builtin_amdgcn_tensor_load_to_lds(g0, g1, gz4, gz4, gz8, 0);
  };

  if (issuer) tdm_issue(0, 0);
  __syncthreads();                               // As staged

  for (int ch = 0; ch < HDIM / BCHUNK; ++ch) {
    const int buf = ch & 1;
    if (issuer) {
      if (ch + 1 < HDIM / BCHUNK) {
        tdm_issue(ch + 1, buf ^ 1);              // prefetch next tile
        __builtin_amdgcn_s_wait_tensorcnt(1);    // current tile landed
      } else {
        __builtin_amdgcn_s_wait_tensorcnt(0);
      }
    }
    __syncthreads();
    const float* bsrc = &Bs[buf][jl * BPAD];
    const int p0 = ch * BCHUNK;
#pragma unroll
    for (int pp = 0; pp < BCHUNK; pp += 4) {
      v2f a, b;
      a.x = As[ml * 516 + p0 + pp + kh];
      a.y = As[ml * 516 + p0 + pp + kh + 1];
      b.x = bsrc[pp + kh];                       // ds_load_b64, conflict-free
      b.y = bsrc[pp + kh + 1];
      acc = wmma4(a, b, acc);
    }
    __syncthreads();
  }
#else
  __syncthreads();
  for (int p = 0; p < HDIM; p += 4) {
    v2f a, b;
    a.x = As[ml * 516 + p + kh];
    a.y = As[ml * 516 + p + kh + 1];
    b.x = W_out[(size_t)j * HDIM + p + kh];
    b.y = W_out[(size_t)j * HDIM + p + kh + 1];
    acc = wmma4(a, b, acc);
  }
#endif

#pragma unroll
  for (int r = 0; r < 8; ++r) {
    int row = m0 + r + ((lane >> 4) << 3);
    out[((size_t)row * KSTEPS + step) * VOCAB + j] = acc[r];
  }
}

// ---------------------------------------------------------------- log-softmax
// In-place over d_out slice `step`; one block per row n.
__global__ __launch_bounds__(256)
void logsoftmax_kernel(float* __restrict__ out, int step) {
  const int n = blockIdx.x;
  float* row = out + ((size_t)n * KSTEPS + step) * VOCAB;
  __shared__ float red[256];

  float mx = -INFINITY;
  for (int v = threadIdx.x; v < VOCAB; v += 256) mx = fmaxf(mx, row[v]);
  red[threadIdx.x] = mx;
  __syncthreads();
  for (int s = 128; s > 0; s >>= 1) {
    if (threadIdx.x < s) red[threadIdx.x] = fmaxf(red[threadIdx.x], red[threadIdx.x + s]);
    __syncthreads();
  }
  mx = red[0];
  __syncthreads();

  float sum = 0.0f;
  for (int v = threadIdx.x; v < VOCAB; v += 256) sum += __expf(row[v] - mx);
  red[threadIdx.x] = sum;
  __syncthreads();
  for (int s = 128; s > 0; s >>= 1) {
    if (threadIdx.x < s) red[threadIdx.x] += red[threadIdx.x + s];
    __syncthreads();
  }
  const float lse = mx + __logf(red[0]);

  for (int v = threadIdx.x; v < VOCAB; v += 256) row[v] = row[v] - lse;
}

// ---------------------------------------------------------------------------
extern "C" void kernel_launch(void* const* d_in, const int* in_sizes, int n_in,
                              void* d_out, int out_size, void* d_ws, size_t ws_size,
                              hipStream_t stream) {
  const float* x         = (const float*)d_in[0];   // embedded_input (N,D)
  // d_in[1] = input_mask (unused by the reference math)
  const float* emb_table = (const float*)d_in[2];   // (V,E)
  const float* W_ih      = (const float*)d_in[3];   // (4H, D+E)
  const float* b_ih      = (const float*)d_in[4];
  const float* W_hh      = (const float*)d_in[5];   // (4H, H)
  const float* b_hh      = (const float*)d_in[6];
  const float* W_out     = (const float*)d_in[7];   // (V, H)
  const float* b_out     = (const float*)d_in[8];
  float* out = (float*)d_out;

  float* ws    = (float*)d_ws;
  float* embl  = ws;                               // N*E     = 131072
  float* gates = embl + N_ROWS * EDIM;             // N*4H    = 1048576
  float* h     = gates + N_ROWS * G4;              // N*H     = 262144
  float* c     = h + N_ROWS * HDIM;                // N*H     = 262144

  // h and c must be zero at the start of every call (deterministic replay)
  zero_kernel<<<(2 * N_ROWS * HDIM + 255) / 256, 256, 0, stream>>>(h, 2 * N_ROWS * HDIM);

  for (int k = 0; k < KSTEPS; ++k) {
    if (k == 0) {
      emb_init_kernel<<<(N_ROWS * EDIM) / 256, 256, 0, stream>>>(emb_table, embl);
    } else {
      emb_bag_kernel<<<dim3(N_ROWS / 16, EDIM / 128), 256, 0, stream>>>(
          out, emb_table, embl, k - 1);
    }
    gates_kernel<<<dim3(N_ROWS / 16, G4 / 128), 256, 0, stream>>>(
        x, embl, h, W_ih, b_ih, W_hh, b_hh, gates);
    cell_kernel<<<(N_ROWS * HDIM) / 256, 256, 0, stream>>>(gates, h, c);
    logits_kernel<<<dim3(N_ROWS / 16, VOCAB / 128), 256, 0, stream>>>(
        h, W_out, b_out, out, k);
    logsoftmax_kernel<<<N_ROWS, 256, 0, stream>>>(out, k);
  }
}